// SAGE_54717883351649
// MI455X (gfx1250) — compile-verified
//
#include <hip/hip_runtime.h>

#define N_NODES 40000
#define E_EDGES 640000
#define DIM     128
#define TM      128                  // rows per block in GEMM
#define WTSTRIDE 260                 // padded dwords per WT row (128 rows x 256 k)
#define ASTRIDE  260                 // padded dwords per Acat row (128 rows x 256 k)
#define WT_FLOATS (128 * WTSTRIDE)   // 33280 floats
#define A_FLOATS  (128 * ASTRIDE)    // 33280 floats
#define LDS_FLOATS (WT_FLOATS + 128 + A_FLOATS)   // ~267KB total

typedef __attribute__((ext_vector_type(2))) float v2f;
typedef __attribute__((ext_vector_type(8))) float v8f;

// ---------------------------------------------------------------- utilities
__global__ void zero_kernel(float* __restrict__ p, int n) {
  int i = blockIdx.x * blockDim.x + threadIdx.x;
  if (i < n) p[i] = 0.0f;
}

__global__ void deg_kernel(const int* __restrict__ dst, float* __restrict__ deg) {
  int e = blockIdx.x * blockDim.x + threadIdx.x;
  if (e < E_EDGES) unsafeAtomicAdd(&deg[dst[e]], 1.0f);
}

// 32 lanes per edge; each lane moves one float4 of the 128-dim feature row
__global__ void scatter_kernel(const float* __restrict__ h,
                               const int* __restrict__ src,
                               const int* __restrict__ dst,
                               float* __restrict__ agg) {
  int idx = blockIdx.x * blockDim.x + threadIdx.x;
  int e = idx >> 5;
  int f = (idx & 31) << 2;
  if (e < E_EDGES) {
    int s = src[e];
    int d = dst[e];
    const float4 v = *(const float4*)(h + (size_t)s * DIM + f);
    float* a = agg + (size_t)d * DIM + f;
    unsafeAtomicAdd(a + 0, v.x);
    unsafeAtomicAdd(a + 1, v.y);
    unsafeAtomicAdd(a + 2, v.z);
    unsafeAtomicAdd(a + 3, v.w);
  }
}

// ------------------------------------------------------- fused SAGE GEMM
// out[m, :] = (agg[m]/max(deg,1)) @ Wl + h[m] @ Wr + b   (optional ReLU)
// Implemented as [A1 | A2](M x 256) @ [Wl ; Wr](256 x 128) with f32 WMMA.
// Weights stored TRANSPOSED in LDS so both WMMA operands are single b64 loads.
__global__ __launch_bounds__(256) void sage_gemm(
    const float* __restrict__ agg, const float* __restrict__ deg,
    const float* __restrict__ h,   const float* __restrict__ Wl,
    const float* __restrict__ Wr,  const float* __restrict__ b,
    float* __restrict__ out, int relu) {
  extern __shared__ float smem[];
  float* WT   = smem;                       // [128 cols][WTSTRIDE] (k fast)
  float* blds = smem + WT_FLOATS;           // [128]
  float* Alds = smem + WT_FLOATS + 128;     // [128 rows][ASTRIDE] (k fast)

  const int tid = threadIdx.x;
  const int rowBase = blockIdx.x * TM;

  // --- stage [Wl ; Wr] transposed into LDS: WT[n][k] = Wcat[k][n]
  for (int i = tid; i < (256 * 128) / 4; i += 256) {
    int base = i << 2;
    int k = base >> 7;          // 0..255
    int n = base & 127;         // 0..124 step 4
    const float* srcw = (k < 128) ? (Wl + k * 128 + n)
                                  : (Wr + (k - 128) * 128 + n);
    float4 v = *(const float4*)srcw;        // coalesced global read
    WT[(n + 0) * WTSTRIDE + k] = v.x;       // transposed LDS writes
    WT[(n + 1) * WTSTRIDE + k] = v.y;
    WT[(n + 2) * WTSTRIDE + k] = v.z;
    WT[(n + 3) * WTSTRIDE + k] = v.w;
  }
  if (tid < 128) blds[tid] = b[tid];

  // --- stage A = [agg/deg | h] tile into LDS, zero-fill past row tail
  for (int i = tid; i < (128 * 256) / 4; i += 256) {
    int base = i << 2;
    int r = base >> 8;
    int c = base & 255;
    int row = rowBase + r;
    float4 v = make_float4(0.f, 0.f, 0.f, 0.f);
    if (row < N_NODES) {
      if (c < 128) {
        v = *(const float4*)(agg + (size_t)row * 128 + c);
        float s = 1.0f / fmaxf(deg[row], 1.0f);
        v.x *= s; v.y *= s; v.z *= s; v.w *= s;
      } else {
        v = *(const float4*)(h + (size_t)row * 128 + (c - 128));
      }
    }
    float* da = Alds + r * ASTRIDE + c;
    da[0] = v.x; da[1] = v.y; da[2] = v.z; da[3] = v.w;
  }
  __syncthreads();

  const int wave = tid >> 5;     // wave handles M-tile `wave` (16 rows)
  const int lane = tid & 31;
  const int half = lane >> 4;    // K sub-pair selector per ISA 16x4 layout
  const int l    = lane & 15;

  v8f acc[8];
#pragma unroll
  for (int nt = 0; nt < 8; ++nt) acc[nt] = (v8f){0, 0, 0, 0, 0, 0, 0, 0};

  const float* arow = Alds + (wave * 16 + l) * ASTRIDE;
  const float* wrow = WT + l * WTSTRIDE;    // + nt*16*WTSTRIDE per N-tile
#pragma unroll 4
  for (int k0 = 0; k0 < 256; k0 += 4) {
    const int kk = k0 + 2 * half;
    v2f a = *(const v2f*)(arow + kk);       // A[M=l][K=kk..kk+1], one b64
#pragma unroll
    for (int nt = 0; nt < 8; ++nt) {
      v2f bb = *(const v2f*)(wrow + nt * 16 * WTSTRIDE + kk);  // one b64
      acc[nt] = __builtin_amdgcn_wmma_f32_16x16x4_f32(
          false, a, false, bb, (short)0, acc[nt], false, false);
    }
  }

  // --- epilogue: bias, optional ReLU, guarded store
#pragma unroll
  for (int nt = 0; nt < 8; ++nt) {
    int col = nt * 16 + l;
    float bias = blds[col];
#pragma unroll
    for (int v = 0; v < 8; ++v) {
      int row = rowBase + wave * 16 + v + 8 * half;
      if (row < N_NODES) {
        float val = acc[nt][v] + bias;
        if (relu) val = fmaxf(val, 0.0f);
        out[(size_t)row * 128 + col] = val;
      }
    }
  }
}

// ---------------------------------------------------------------- launcher
extern "C" void kernel_launch(void* const* d_in, const int* in_sizes, int n_in,
                              void* d_out, int out_size, void* d_ws, size_t ws_size,
                              hipStream_t stream) {
  const float* x   = (const float*)d_in[0];
  const int*  edge = (const int*)d_in[1];   // [2, E]
  const int*  src  = edge;
  const int*  dst  = edge + E_EDGES;
  const float* Wl[3] = {(const float*)d_in[2], (const float*)d_in[5], (const float*)d_in[8]};
  const float* Wr[3] = {(const float*)d_in[3], (const float*)d_in[6], (const float*)d_in[9]};
  const float* bs[3] = {(const float*)d_in[4], (const float*)d_in[7], (const float*)d_in[10]};

  float* bufA = (float*)d_ws;
  float* bufB = bufA + (size_t)N_NODES * DIM;
  float* agg  = bufB + (size_t)N_NODES * DIM;
  float* deg  = agg  + (size_t)N_NODES * DIM;

  // degree (same for all layers; recomputed each call for determinism)
  zero_kernel<<<(N_NODES + 255) / 256, 256, 0, stream>>>(deg, N_NODES);
  deg_kernel<<<(E_EDGES + 255) / 256, 256, 0, stream>>>(dst, deg);

  const float* hin = x;
  float* houts[3] = {bufA, bufB, (float*)d_out};
  const int nAgg = N_NODES * DIM;
  const size_t ldsBytes = (size_t)LDS_FLOATS * sizeof(float);
  const int gemmGrid = (N_NODES + TM - 1) / TM;

  for (int layer = 0; layer < 3; ++layer) {
    zero_kernel<<<(nAgg + 255) / 256, 256, 0, stream>>>(agg, nAgg);
    scatter_kernel<<<(E_EDGES * 32 + 255) / 256, 256, 0, stream>>>(hin, src, dst, agg);
    sage_gemm<<<gemmGrid, 256, ldsBytes, stream>>>(
        agg, deg, hin, Wl[layer], Wr[layer], bs[layer], houts[layer],
        layer != 2 ? 1 : 0);
    hin = houts[layer];
  }
}